// BackWarp_35158602285813
// MI455X (gfx1250) — compile-verified
//
#include <hip/hip_runtime.h>

// Dense image warp (backward bilinear warp), matching the JAX reference:
//   query = (y,x) - flow;  f = clamp(floor(q), 0, size-2);  a = clamp(q-f, 0, 1)
//   out = lerp_y(lerp_x(tl,tr,ax), lerp_x(bl,br,ax), ay)
//
// Shapes fixed by the reference: image [16,720,1280,3] f32, flow [16,720,1280,2] f32.

#define BB 16
#define HH 720
#define WW 1280
#define CC 3

typedef float v2f __attribute__((ext_vector_type(2)));

__global__ __launch_bounds__(256) void backwarp_kernel(
    const float* __restrict__ img,   // [B,H,W,3]
    const float* __restrict__ flow,  // [B,H,W,2]
    float* __restrict__ out)         // [B,H,W,3]
{
    const int x = blockIdx.x * 256 + threadIdx.x;   // W = 1280 = 5 * 256, always in-bounds
    const int y = blockIdx.y;
    const int b = blockIdx.z;

    const long long pix = ((long long)b * HH + y) * WW + x;

    // Flow is streamed exactly once -> nontemporal 8B coalesced load.
    v2f f = __builtin_nontemporal_load((const v2f*)flow + pix);

    // query = grid - flow   (flow channel 0 is y, channel 1 is x)
    const float qy = (float)y - f.x;
    const float qx = (float)x - f.y;

    // floor clamped to [0, size-2]; alpha clamped to [0,1]  (reference semantics)
    float fy = floorf(qy);
    float fx = floorf(qx);
    fy = fminf(fmaxf(fy, 0.0f), (float)(HH - 2));
    fx = fminf(fmaxf(fx, 0.0f), (float)(WW - 2));
    const float ay = fminf(fmaxf(qy - fy, 0.0f), 1.0f);
    const float ax = fminf(fmaxf(qx - fx, 0.0f), 1.0f);
    const int iy = (int)fy;
    const int ix = (int)fx;

    // Gather: top pair (tl,tr) is 6 contiguous floats, bottom pair likewise.
    const float* __restrict__ top = img + (long long)b * (HH * (long long)WW * CC)
                                        + ((long long)iy * WW + ix) * CC;
    const float* __restrict__ bot = top + (long long)WW * CC;

    const float tl0 = top[0], tl1 = top[1], tl2 = top[2];
    const float tr0 = top[3], tr1 = top[4], tr2 = top[5];
    const float bl0 = bot[0], bl1 = bot[1], bl2 = bot[2];
    const float br0 = bot[3], br1 = bot[4], br2 = bot[5];

    // interp_top = ax*(tr-tl)+tl ; interp_bottom = ax*(br-bl)+bl ; out = ay*(ib-it)+it
    const float it0 = fmaf(ax, tr0 - tl0, tl0);
    const float it1 = fmaf(ax, tr1 - tl1, tl1);
    const float it2 = fmaf(ax, tr2 - tl2, tl2);
    const float ib0 = fmaf(ax, br0 - bl0, bl0);
    const float ib1 = fmaf(ax, br1 - bl1, bl1);
    const float ib2 = fmaf(ax, br2 - bl2, bl2);

    float* __restrict__ op = out + pix * CC;   // contiguous 12B per lane -> b96 store
    op[0] = fmaf(ay, ib0 - it0, it0);
    op[1] = fmaf(ay, ib1 - it1, it1);
    op[2] = fmaf(ay, ib2 - it2, it2);
}

extern "C" void kernel_launch(void* const* d_in, const int* in_sizes, int n_in,
                              void* d_out, int out_size, void* d_ws, size_t ws_size,
                              hipStream_t stream) {
    (void)in_sizes; (void)n_in; (void)out_size; (void)d_ws; (void)ws_size;
    const float* img  = (const float*)d_in[0];
    const float* flow = (const float*)d_in[1];
    float* out = (float*)d_out;

    dim3 block(256, 1, 1);
    dim3 grid(WW / 256, HH, BB);   // (5, 720, 16)
    backwarp_kernel<<<grid, block, 0, stream>>>(img, flow, out);
}